// MultiHeadAttentionQuantum_65481071408524
// MI455X (gfx1250) — compile-verified
//
#include <hip/hip_runtime.h>

// Fused: out = cumprod8(cos(x @ Wq^T)) @ Wo^T   (k/v GEMMs are dead compute)
// B=8, S=4096, E=512 -> M = 32768 rows.
// All-fp32 via V_WMMA_F32_16X16X4_F32; 2x2 register tiling per wave to halve
// the load-per-WMMA ratio (A reused across N-tiles, B reused across M-tiles).

typedef __attribute__((ext_vector_type(2))) float v2f;
typedef __attribute__((ext_vector_type(8))) float v8f;

#define E_DIM 512
#define BM 32          // rows per workgroup
#define NTHREADS 256   // 8 wave32 per workgroup

__global__ __launch_bounds__(NTHREADS) void fused_qmha_kernel(
    const float* __restrict__ x,
    const float* __restrict__ Wq,
    const float* __restrict__ Wo,
    float* __restrict__ out)
{
    // 32 x 512 fp32 tile of q/meas, XOR-swizzled columns -> 64 KB LDS.
    __shared__ float qs[BM * E_DIM];

    const int rowBase = blockIdx.x * BM;
    const int lane = threadIdx.x & 31;
    const int wave = threadIdx.x >> 5;
    const int l = lane & 15;   // row (A) / column (B,C,D) index within tile
    const int h = lane >> 4;   // K-pair / M-half selector

    // ---------------- GEMM1: q = x @ Wq^T ----------------
    // Wave w owns nt in {4w..4w+3} x mt in {0,1}; two 2x2 tile blocks.
    const float* a0p = x + (size_t)(rowBase + l) * E_DIM + 2 * h;        // mt=0
    const float* a1p = a0p + (size_t)16 * E_DIM;                          // mt=1

    for (int j = 0; j < 2; ++j) {
        const int nt0 = wave * 4 + 2 * j;
        const float* b0p = Wq + (size_t)(nt0 * 16 + l) * E_DIM + 2 * h;  // nt0
        const float* b1p = b0p + (size_t)16 * E_DIM;                      // nt0+1
        v8f acc00 = {}, acc01 = {}, acc10 = {}, acc11 = {};
        #pragma unroll 4
        for (int k0 = 0; k0 < E_DIM; k0 += 4) {
            v2f a0 = *(const v2f*)(a0p + k0);
            v2f a1 = *(const v2f*)(a1p + k0);
            v2f b0 = *(const v2f*)(b0p + k0);
            v2f b1 = *(const v2f*)(b1p + k0);
            acc00 = __builtin_amdgcn_wmma_f32_16x16x4_f32(false, a0, false, b0, (short)0, acc00, false, false);
            acc01 = __builtin_amdgcn_wmma_f32_16x16x4_f32(false, a0, false, b1, (short)0, acc01, false, false);
            acc10 = __builtin_amdgcn_wmma_f32_16x16x4_f32(false, a1, false, b0, (short)0, acc10, false, false);
            acc11 = __builtin_amdgcn_wmma_f32_16x16x4_f32(false, a1, false, b1, (short)0, acc11, false, false);
        }
        // D-frags -> LDS with XOR bank swizzle; rows r and r+16 share the swizzle.
        #pragma unroll
        for (int p = 0; p < 8; ++p) {
            const int r  = p + 8 * h;            // mt=0 tile-local row (0..15)
            const int sw = r * 4;
            const int c0 = nt0 * 16 + l;
            qs[r * E_DIM + (c0 ^ sw)]                       = acc00[p];
            qs[r * E_DIM + ((c0 + 16) ^ sw)]                = acc01[p];
            qs[(r + 16) * E_DIM + (c0 ^ sw)]                = acc10[p];
            qs[(r + 16) * E_DIM + ((c0 + 16) ^ sw)]         = acc11[p];
        }
    }
    __syncthreads();

    // ---------------- cos + cumprod over each 8-wide wire group ----------------
    for (int g = threadIdx.x; g < BM * E_DIM / 8; g += NTHREADS) {
        const int r  = g >> 6;          // 0..31
        const int c0 = (g & 63) * 8;    // group base column
        const int s  = (r & 15) * 4;    // swizzle for this row
        float run = 1.0f;
        #pragma unroll
        for (int jj = 0; jj < 8; ++jj) {
            const int idx = r * E_DIM + ((c0 + jj) ^ s);
            run *= cosf(qs[idx]);
            qs[idx] = run;              // meas[j] = prod_{i<=j} cos(q[i])
        }
    }
    __syncthreads();

    // ---------------- GEMM2: out = meas @ Wo^T ----------------
    const int swA = l * 4;             // swizzle for LDS A rows l and l+16
    for (int j = 0; j < 2; ++j) {
        const int nt0 = wave * 4 + 2 * j;
        const float* b0p = Wo + (size_t)(nt0 * 16 + l) * E_DIM + 2 * h;
        const float* b1p = b0p + (size_t)16 * E_DIM;
        v8f acc00 = {}, acc01 = {}, acc10 = {}, acc11 = {};
        #pragma unroll 4
        for (int k0 = 0; k0 < E_DIM; k0 += 4) {
            const int cc = (k0 + 2 * h) ^ swA;   // even -> 8B-aligned ds_load_b64
            v2f a0 = *(const v2f*)(&qs[l * E_DIM + cc]);
            v2f a1 = *(const v2f*)(&qs[(16 + l) * E_DIM + cc]);
            v2f b0 = *(const v2f*)(b0p + k0);
            v2f b1 = *(const v2f*)(b1p + k0);
            acc00 = __builtin_amdgcn_wmma_f32_16x16x4_f32(false, a0, false, b0, (short)0, acc00, false, false);
            acc01 = __builtin_amdgcn_wmma_f32_16x16x4_f32(false, a0, false, b1, (short)0, acc01, false, false);
            acc10 = __builtin_amdgcn_wmma_f32_16x16x4_f32(false, a1, false, b0, (short)0, acc10, false, false);
            acc11 = __builtin_amdgcn_wmma_f32_16x16x4_f32(false, a1, false, b1, (short)0, acc11, false, false);
        }
        #pragma unroll
        for (int p = 0; p < 8; ++p) {
            const size_t r0 = (size_t)(rowBase + p + 8 * h);
            const size_t r1 = r0 + 16;
            const int    c0 = nt0 * 16 + l;
            out[r0 * E_DIM + c0]      = acc00[p];
            out[r0 * E_DIM + c0 + 16] = acc01[p];
            out[r1 * E_DIM + c0]      = acc10[p];
            out[r1 * E_DIM + c0 + 16] = acc11[p];
        }
    }
}

extern "C" void kernel_launch(void* const* d_in, const int* in_sizes, int n_in,
                              void* d_out, int out_size, void* d_ws, size_t ws_size,
                              hipStream_t stream) {
    (void)n_in; (void)out_size; (void)d_ws; (void)ws_size;
    const float* x  = (const float*)d_in[0];
    const float* Wq = (const float*)d_in[1];
    // d_in[2] (Wk) and d_in[3] (Wv) feed dead compute in the reference -> skipped.
    const float* Wo = (const float*)d_in[4];
    float* out = (float*)d_out;

    const int M = in_sizes[0] / E_DIM;       // B*S = 32768
    dim3 grid(M / BM), block(NTHREADS);
    fused_qmha_kernel<<<grid, block, 0, stream>>>(x, Wq, Wo, out);
}